// ECELoss_62517543960687
// MI455X (gfx1250) — compile-verified
//
#include <hip/hip_runtime.h>
#include <stdint.h>

#define N_BINS 15
#define C 128
#define WAVES_PER_BLOCK 8
#define BLOCK 256

// ---------------------------------------------------------------------------
// Zero the 45-float workspace (counts[15], conf_sum[15], acc_sum[15]).
// Launched every call so graph replays are deterministic.
// ---------------------------------------------------------------------------
__global__ void ece_zero_ws(float* __restrict__ ws) {
    int t = threadIdx.x;
    if (t < 3 * N_BINS) ws[t] = 0.0f;
}

// ---------------------------------------------------------------------------
// Main kernel: one wave32 per row, rows double-buffered through LDS using the
// CDNA5 async global->LDS DMA path (ASYNCcnt), non-temporal since logits are
// streamed exactly once. Each lane owns a 16B chunk of the 512B row.
// Wave shuffle reductions -> max/argmax and exp-sum. Per-block LDS bin
// accumulators (ds_add_f32), flushed once via global_atomic_add_f32.
// ---------------------------------------------------------------------------
__global__ __launch_bounds__(BLOCK) void ece_main(const float* __restrict__ logits,
                                                  const int* __restrict__ labels,
                                                  float* __restrict__ ws,
                                                  int N) {
    __shared__ float bins[3 * N_BINS];
    __shared__ __align__(16) float buf[WAVES_PER_BLOCK][2][C];

    const int tid = threadIdx.x;
    for (int i = tid; i < 3 * N_BINS; i += BLOCK) bins[i] = 0.0f;
    __syncthreads();

    const int lane = tid & 31;
    const int wid  = tid >> 5;
    int row = blockIdx.x * WAVES_PER_BLOCK + wid;          // N = 2M -> fits i32
    const int wstride = gridDim.x * WAVES_PER_BLOCK;

    // LDS byte addresses of this wave's two per-lane 16B slots.
    // (Low 32 bits of a generic pointer into __shared__ == LDS byte offset.)
    const uint32_t lds0 = (uint32_t)(uintptr_t)(&buf[wid][0][lane * 4]);
    const uint32_t lds1 = (uint32_t)(uintptr_t)(&buf[wid][1][lane * 4]);

    // Incremental 64-bit addressing: one add per iteration instead of a
    // 64-bit multiply + adds (cuts hot-loop SALU/VALU).
    const uint64_t step = (uint64_t)wstride * (C * 4);
    uint64_t ga = (uint64_t)(uintptr_t)logits + (uint64_t)row * (C * 4)
                + (uint32_t)(lane * 16);

    if (row < N) {
        asm volatile("global_load_async_to_lds_b128 %0, %1, off th:TH_LOAD_NT"
                     :: "v"(lds0), "v"(ga) : "memory");
    }

    int sel = 0;
    for (; row < N; row += wstride) {
        // Wait for the current buffer's async DMA to land in LDS.
        asm volatile("s_wait_asynccnt 0x0" ::: "memory");
        // Retire previous iteration's ds_loads from the buffer we are about to
        // refill (DScnt and ASYNCcnt are unordered in hardware).
        asm volatile("s_wait_dscnt 0x0" ::: "memory");

        // Prefetch the next row into the other buffer; overlaps with compute.
        if (row + wstride < N) {
            uint64_t gan  = ga + step;
            uint32_t ldsn = sel ? lds0 : lds1;
            asm volatile("global_load_async_to_lds_b128 %0, %1, off th:TH_LOAD_NT"
                         :: "v"(ldsn), "v"(gan) : "memory");
        }

        // ds_load_b128: this lane's 4 contiguous logits of the current row.
        const float4 v = *reinterpret_cast<const float4*>(&buf[wid][sel][lane * 4]);

        // Lane-local max + argmax (first-index tiebreak).
        float m = v.x; int mi = 0;
        if (v.y > m) { m = v.y; mi = 1; }
        if (v.z > m) { m = v.z; mi = 2; }
        if (v.w > m) { m = v.w; mi = 3; }
        int idx = lane * 4 + mi;

        // Wave32 reduction: (max, first argmax).
        #pragma unroll
        for (int off = 16; off; off >>= 1) {
            float om = __shfl_xor(m, off, 32);
            int   oi = __shfl_xor(idx, off, 32);
            if (om > m || (om == m && oi < idx)) { m = om; idx = oi; }
        }

        // Sum of exp(x - m) across the row (4x v_exp_f32 per lane).
        float s = __expf(v.x - m) + __expf(v.y - m) +
                  __expf(v.z - m) + __expf(v.w - m);
        #pragma unroll
        for (int off = 16; off; off >>= 1) s += __shfl_xor(s, off, 32);

        if (lane == 0) {
            float conf = 1.0f / s;                 // exp(m-m)/sum = max softmax
            float acc  = (idx == labels[row]) ? 1.0f : 0.0f;
            int b = (int)ceilf(conf * (float)N_BINS) - 1;
            b = b < 0 ? 0 : (b > N_BINS - 1 ? N_BINS - 1 : b);
            atomicAdd(&bins[b],              1.0f);   // ds_add_f32
            atomicAdd(&bins[N_BINS + b],     conf);
            atomicAdd(&bins[2 * N_BINS + b], acc);
        }

        sel ^= 1;
        ga  += step;
    }

    __syncthreads();
    if (tid < 3 * N_BINS) atomicAdd(&ws[tid], bins[tid]);  // global_atomic_add_f32
}

// ---------------------------------------------------------------------------
// Finalize: one wave reduces the 15 bins into the ECE scalar.
// ---------------------------------------------------------------------------
__global__ void ece_final(const float* __restrict__ ws, float* __restrict__ out,
                          int N) {
    const int lane = threadIdx.x;
    float term = 0.0f;
    if (lane < N_BINS) {
        float cnt = ws[lane];
        if (cnt > 0.0f) {
            float avg_conf = ws[N_BINS + lane] / cnt;
            float avg_acc  = ws[2 * N_BINS + lane] / cnt;
            term = fabsf(avg_conf - avg_acc) * (cnt / (float)N);
        }
    }
    #pragma unroll
    for (int off = 16; off; off >>= 1) term += __shfl_xor(term, off, 32);
    if (lane == 0) out[0] = term;
}

// ---------------------------------------------------------------------------
extern "C" void kernel_launch(void* const* d_in, const int* in_sizes, int n_in,
                              void* d_out, int out_size, void* d_ws, size_t ws_size,
                              hipStream_t stream) {
    const float* logits = (const float*)d_in[0];
    const int*   labels = (const int*)d_in[1];
    float*       ws     = (float*)d_ws;
    float*       out    = (float*)d_out;

    const int N = in_sizes[1];   // labels count == number of rows

    ece_zero_ws<<<1, 64, 0, stream>>>(ws);

    // 3072 blocks * 8 waves = 24,576 waves; each wave keeps one 512B async DMA
    // in flight -> ~12.6 MB of outstanding reads, ample to saturate 23.3 TB/s.
    int blocks = 3072;
    long long waves_needed = ((long long)N + WAVES_PER_BLOCK - 1) / WAVES_PER_BLOCK;
    if ((long long)blocks > waves_needed) blocks = (int)waves_needed;
    if (blocks < 1) blocks = 1;

    ece_main<<<blocks, BLOCK, 0, stream>>>(logits, labels, ws, N);
    ece_final<<<1, 32, 0, stream>>>(ws, out, N);
}